// SPDStemEdgeEncoder_66271345377500
// MI455X (gfx1250) — compile-verified
//
#include <hip/hip_runtime.h>
#include <hip/hip_bf16.h>
#include <stdint.h>

#define N_NODES 50000
#define E_SPD   2000000
#define E_EDGE  500000
#define OUT_DIM 64
#define RBLOCKS 1024   // blocks in reduction kernel (256 thr = 8 waves each)

typedef __attribute__((ext_vector_type(2))) float v2f;
typedef __attribute__((ext_vector_type(8))) float v8f;

// ---------------------------------------------------------------- helpers ---
__device__ __forceinline__ v8f wmma_colsum_acc(v2f a, v2f b, v8f c) {
  // D = A(16x4) * B(4x16) + C ; with A == all-ones every row of D is the
  // K-wise column sum of B. 8-arg form: (neg_a, A, neg_b, B, c_mod, C, ra, rb)
  return __builtin_amdgcn_wmma_f32_16x16x4_f32(false, a, false, b, (short)0, c,
                                               false, false);
}

__device__ __forceinline__ void async_b128_to_lds(uint32_t lds_byte_addr,
                                                  const void* gaddr) {
  // CDNA5 async DMA: global -> LDS, tracked by ASYNCcnt.
  asm volatile("global_load_async_to_lds_b128 %0, %1, off"
               :: "v"(lds_byte_addr), "v"(gaddr)
               : "memory");
}

__device__ __forceinline__ void wait_asynccnt0() {
  asm volatile("s_wait_asynccnt 0" ::: "memory");
}

// ------------------------------------------------------------- 1) gather ----
// out_val[row, :] = pe_emb[spd_val[row], :]
// pe_emb (17*64 f32 = 4352 B) is staged into LDS with async b128 DMA.
__global__ void gather_kernel(const int* __restrict__ spd_val,
                              const float* __restrict__ pe_emb,
                              float4* __restrict__ out) {
  __shared__ __align__(16) float tab[17 * OUT_DIM];
  const int t = threadIdx.x;

  const uint32_t lds0 = (uint32_t)(uintptr_t)&tab[0];
  const char* g = (const char*)pe_emb;
  async_b128_to_lds(lds0 + t * 16, g + t * 16);            // 4096 B
  if (t < 16)
    async_b128_to_lds(lds0 + 4096 + t * 16, g + 4096 + t * 16); // last 256 B
  wait_asynccnt0();
  __syncthreads();

  const int idx = blockIdx.x * 256 + t;        // one float4 per thread
  if (idx < E_SPD * 16) {
    const int row = idx >> 4;
    const int c4  = idx & 15;
    const int v   = spd_val[row];              // 0..16
    const float4* tp = (const float4*)tab;
    out[idx] = tp[v * 16 + c4];
  }
}

// --------------------------------------------------- 2) edge scatter-add ----
// pos = bisect_left over lin_spd (int32 wrap, matching JAX x64-off semantics)
__global__ void scatter_kernel(const int* __restrict__ spd_index,
                               const int* __restrict__ edge_index,
                               const float* __restrict__ edge_attr,
                               float* __restrict__ out) {
  __shared__ int posb[16];
  const int t = threadIdx.x;
  const int ebase = blockIdx.x * 16;

  if (t < 16) {
    const int e  = ebase + t;
    const int e0 = edge_index[e];
    const int e1 = edge_index[E_EDGE + e];
    const int key = (int)((unsigned)e0 * (unsigned)N_NODES + (unsigned)e1);
    int lo = 0, hi = E_SPD;
    while (lo < hi) {
      const int mid = (lo + hi) >> 1;
      const int m0 = spd_index[mid];
      const int m1 = spd_index[E_SPD + mid];
      const int mv = (int)((unsigned)m0 * (unsigned)N_NODES + (unsigned)m1);
      if (mv < key) lo = mid + 1; else hi = mid;
    }
    posb[t] = lo;
  }
  __syncthreads();

  const int el  = t >> 4;
  const int c4  = t & 15;
  const int pos = posb[el];
  if (pos < E_SPD) {                                  // JAX drops OOB scatter
    const float4 a = ((const float4*)edge_attr)[(ebase + el) * 16 + c4];
    float* dst = out + (long long)pos * OUT_DIM + c4 * 4;
    atomicAdd(dst + 0, a.x);
    atomicAdd(dst + 1, a.y);
    atomicAdd(dst + 2, a.z);
    atomicAdd(dst + 3, a.w);
  }
}

// --------------------------------------- 3) column sum / sumsq via WMMA ----
// Per wave: tile of 4 rows. B(4x16) per 16-col chunk; A = ones => WMMA
// accumulates column sums. Squares go through a second accumulator.
__global__ void reduce_kernel(const float* __restrict__ out,
                              float* __restrict__ partials) {
  const int t = threadIdx.x;
  const int lane = t & 31;
  const int waveInBlock = t >> 5;                      // 0..7
  const int wave  = blockIdx.x * 8 + waveInBlock;
  const int nWaves = RBLOCKS * 8;

  v8f csum[4], csq[4];
#pragma unroll
  for (int ch = 0; ch < 4; ++ch) { csum[ch] = (v8f)0.0f; csq[ch] = (v8f)0.0f; }

  const v2f ones = {1.0f, 1.0f};
  const int colBase = lane & 15;
  const int rsel = (lane < 16) ? 0 : 2;                // rows 0,1 | 2,3 of tile

  const int NTILES = E_SPD / 4;
  for (int tile = wave; tile < NTILES; tile += nWaves) {   // wave-uniform
    const float* p = out + (long long)(tile * 4 + rsel) * OUT_DIM;
#pragma unroll
    for (int ch = 0; ch < 4; ++ch) {
      const int col = ch * 16 + colBase;
      v2f b;
      b.x = p[col];
      b.y = p[OUT_DIM + col];
      csum[ch] = wmma_colsum_acc(ones, b, csum[ch]);
      v2f b2 = b * b;
      csq[ch] = wmma_colsum_acc(ones, b2, csq[ch]);
    }
  }

  // D row M=0 lives in VGPR0 lanes 0..15 (N = lane). Fixed-order combine.
  __shared__ float sh[8][128];
  if (lane < 16) {
#pragma unroll
    for (int ch = 0; ch < 4; ++ch) {
      sh[waveInBlock][ch * 16 + lane]      = csum[ch][0];
      sh[waveInBlock][64 + ch * 16 + lane] = csq[ch][0];
    }
  }
  __syncthreads();
  if (t < 128) {
    float s = 0.0f;
#pragma unroll
    for (int w = 0; w < 8; ++w) s += sh[w][t];
    partials[blockIdx.x * 128 + t] = s;
  }
}

// ------------------------------------------------- 4) finalize BN params ----
__global__ void finalize_kernel(const float* __restrict__ partials,
                                const float* __restrict__ gamma,
                                const float* __restrict__ beta,
                                float* __restrict__ scale_shift) {
  const int t = threadIdx.x;                           // 0..127
  float s = 0.0f;
  for (int b = 0; b < RBLOCKS; ++b) s += partials[b * 128 + t];
  __shared__ float acc[128];
  acc[t] = s;
  __syncthreads();
  if (t < 64) {
    const float n    = (float)E_SPD;
    const float mean = acc[t] / n;
    float var        = acc[64 + t] / n - mean * mean;
    var = var < 0.0f ? 0.0f : var;
    const float rs = rsqrtf(var + 1e-5f);
    const float sc = rs * gamma[t];
    scale_shift[t]      = sc;
    scale_shift[64 + t] = beta[t] - mean * sc;
  }
}

// ----------------------------------------------------- 5) normalize rows ----
__global__ void normalize_kernel(float4* __restrict__ out,
                                 const float* __restrict__ scale_shift) {
  __shared__ float4 ssc[16], ssh[16];
  const int t = threadIdx.x;
  if (t < 16) {
    ssc[t] = ((const float4*)scale_shift)[t];
    ssh[t] = ((const float4*)(scale_shift + 64))[t];
  }
  __syncthreads();

  const int idx = blockIdx.x * 256 + t;
  if (idx < E_SPD * 16) {
    const int c4 = idx & 15;
    float4 x = out[idx];
    const float4 sc = ssc[c4];
    const float4 sh = ssh[c4];
    x.x = fmaf(x.x, sc.x, sh.x);
    x.y = fmaf(x.y, sc.y, sh.y);
    x.z = fmaf(x.z, sc.z, sh.z);
    x.w = fmaf(x.w, sc.w, sh.w);
    out[idx] = x;
  }
}

// ------------------------------------------------------------------ host ----
extern "C" void kernel_launch(void* const* d_in, const int* in_sizes, int n_in,
                              void* d_out, int out_size, void* d_ws,
                              size_t ws_size, hipStream_t stream) {
  const int*   spd_index = (const int*)d_in[0];   // (2, E_SPD) int32
  const int*   spd_val   = (const int*)d_in[1];   // (E_SPD)   int32
  const int*   edge_index= (const int*)d_in[2];   // (2, E_EDGE) int32
  const float* edge_attr = (const float*)d_in[3]; // (E_EDGE, 64) f32
  const float* pe_emb    = (const float*)d_in[4]; // (17, 64) f32
  const float* bn_gamma  = (const float*)d_in[5]; // (64) f32
  const float* bn_beta   = (const float*)d_in[6]; // (64) f32
  (void)in_sizes; (void)n_in; (void)out_size;

  // Output layout: [spd_index (4M int32 words)] [out_val (128M f32)]
  float* out_val = (float*)d_out + 2 * (size_t)E_SPD;

  // Workspace layout (floats): [block partials RBLOCKS*128][scale 64][shift 64]
  float* partials    = (float*)d_ws;
  float* scale_shift = partials + RBLOCKS * 128;
  (void)ws_size;

  // 0) pass-through first output
  hipMemcpyAsync(d_out, d_in[0], 2 * (size_t)E_SPD * sizeof(int),
                 hipMemcpyDeviceToDevice, stream);

  // 1) gather (overwrites whole out_val region every call -> deterministic)
  const int nF4 = E_SPD * 16;                     // 32M float4 slots
  gather_kernel<<<nF4 / 256, 256, 0, stream>>>(spd_val, pe_emb,
                                               (float4*)out_val);

  // 2) edge scatter-add
  scatter_kernel<<<E_EDGE / 16, 256, 0, stream>>>(spd_index, edge_index,
                                                  edge_attr, out_val);

  // 3) WMMA column reduction -> per-block partials
  reduce_kernel<<<RBLOCKS, 256, 0, stream>>>(out_val, partials);

  // 4) finalize mean/var -> scale/shift
  finalize_kernel<<<1, 128, 0, stream>>>(partials, bn_gamma, bn_beta,
                                         scale_shift);

  // 5) in-place batch-norm
  normalize_kernel<<<nF4 / 256, 256, 0, stream>>>((float4*)out_val,
                                                  scale_shift);
}